// GAT_57363583205415
// MI455X (gfx1250) — compile-verified
//
#include <hip/hip_runtime.h>
#include <hip/hip_bf16.h>

// ---------------------------------------------------------------------------
// GATv2 (3 layers) + graph heads + MLP on MI455X (gfx1250, wave32).
// Node GEMMs use V_WMMA_F32_16X16X4_F32 (full f32 precision, matches ref).
// Edge work is L2-resident gather (h = 25.6MB << 192MB L2) + f32 atomics.
// ---------------------------------------------------------------------------

#define NNODES 50000
#define NEDGES 800000
#define NGRAPH 64
#define HD 128           // heads*feats = 4*32
#define NHEAD 4
#define HID 32
#define NOUT 10

typedef __attribute__((ext_vector_type(2))) float v2f;
typedef __attribute__((ext_vector_type(8))) float v8f;

// ---------------------------------------------------------------------------
// h = x @ W + fb   ([N,128] x [128,128]).  One wave computes a 16x128 slab:
// 8 accumulators (16x16 f32 tiles), K swept in steps of 4 via wmma_f32_16x16x4.
// A frag (16x4 f32): lanes 0-15 carry K pair {0,1}, lanes 16-31 pair {2,3}.
// B frag (4x16 f32): mirrored; C/D: vgpr r -> M=r (lanes 0-15), M=r+8 (16-31).
// ---------------------------------------------------------------------------
__global__ __launch_bounds__(256)
void gemm_node_wmma(const float* __restrict__ X, const float* __restrict__ W,
                    const float* __restrict__ bias, float* __restrict__ Hout,
                    int nrows)
{
    const int lane = threadIdx.x & 31;
    const int wave = threadIdx.x >> 5;
    const int tile = blockIdx.x * 8 + wave;      // 16-row tile index
    const int row0 = tile * 16;
    if (row0 >= nrows) return;                   // wave-uniform: WMMA waves keep EXEC all-1s

    const int m  = lane & 15;                    // row within tile (A) / col within tile (B,D)
    const int kh = lane >> 4;                    // which K pair of 4

    v8f acc[8] = {};
    const float* xrow = X + (size_t)(row0 + m) * HD;

    for (int k0 = 0; k0 < HD; k0 += 4) {
        const int ka = k0 + 2 * kh;
        v2f a;
        a.x = xrow[ka];
        a.y = xrow[ka + 1];
#pragma unroll
        for (int t = 0; t < 8; ++t) {
            const int col = t * 16 + m;
            v2f b;
            b.x = W[(size_t)ka * HD + col];
            b.y = W[(size_t)(ka + 1) * HD + col];
            acc[t] = __builtin_amdgcn_wmma_f32_16x16x4_f32(
                false, a, false, b, (short)0, acc[t], false, false);
        }
    }

#pragma unroll
    for (int t = 0; t < 8; ++t) {
        const int col = t * 16 + m;
        const float bc = bias[col];
#pragma unroll
        for (int r = 0; r < 8; ++r) {
            const int row = row0 + r + 8 * kh;
            Hout[(size_t)row * HD + col] = acc[t][r] + bc;
        }
    }
}

// ---------------------------------------------------------------------------
// Pass A over edges: e[h] = sum_f leaky(h[src]+h[dst])[h,f]*attn[h,f];
// ex = exp(e) (max-shift dropped: softmax invariant, scores are O(1));
// den[dst,h] += ex.   32 lanes per edge, lane f covers feature f of each head.
// ---------------------------------------------------------------------------
__global__ __launch_bounds__(256)
void edge_scores(const float* __restrict__ Hfeat, const int* __restrict__ src,
                 const int* __restrict__ dst, const float* __restrict__ attn,
                 float* __restrict__ ex, float* __restrict__ den, int nedges)
{
    const int gid  = blockIdx.x * blockDim.x + threadIdx.x;
    const int e    = gid >> 5;
    const int lane = gid & 31;
    if (e >= nedges) return;
    const int s = src[e], d = dst[e];
    const float* hs = Hfeat + (size_t)s * HD;
    const float* hd = Hfeat + (size_t)d * HD;

    float p[NHEAD];
#pragma unroll
    for (int h = 0; h < NHEAD; ++h) {
        const int c = h * 32 + lane;
        float v = hs[c] + hd[c];
        v = (v > 0.0f) ? v : 0.2f * v;           // leaky_relu slope 0.2
        p[h] = v * attn[c];
    }
#pragma unroll
    for (int off = 16; off > 0; off >>= 1) {
#pragma unroll
        for (int h = 0; h < NHEAD; ++h) p[h] += __shfl_xor(p[h], off, 32);
    }
    if (lane == 0) {
#pragma unroll
        for (int h = 0; h < NHEAD; ++h) {
            const float x = __expf(p[h]);
            ex[(size_t)e * NHEAD + h] = x;
            atomicAdd(&den[(size_t)d * NHEAD + h], x);
        }
    }
}

// ---------------------------------------------------------------------------
// Pass B over edges: alpha = ex/den[dst]; agg[dst] += h[src] * alpha[head].
// ---------------------------------------------------------------------------
__global__ __launch_bounds__(256)
void edge_aggregate(const float* __restrict__ Hfeat, const int* __restrict__ src,
                    const int* __restrict__ dst, const float* __restrict__ ex,
                    const float* __restrict__ den, float* __restrict__ agg,
                    int nedges)
{
    const int gid  = blockIdx.x * blockDim.x + threadIdx.x;
    const int e    = gid >> 5;
    const int lane = gid & 31;
    if (e >= nedges) return;
    const int s = src[e], d = dst[e];

    float alpha[NHEAD];
#pragma unroll
    for (int h = 0; h < NHEAD; ++h) {
        const float dn = den[(size_t)d * NHEAD + h];
        alpha[h] = ex[(size_t)e * NHEAD + h] / ((dn > 0.0f) ? dn : 1.0f);
    }
    const float* hs = Hfeat + (size_t)s * HD;
#pragma unroll
    for (int h = 0; h < NHEAD; ++h) {
        const int c = h * 32 + lane;
        atomicAdd(&agg[(size_t)d * HD + c], hs[c] * alpha[h]);
    }
}

// ---------------------------------------------------------------------------
// x = relu(agg + ob); accumulate per-column sum / sumsq for BN (biased var).
// 65536 threads, thread -> fixed column, strided rows, two atomics per thread.
// ---------------------------------------------------------------------------
__global__ __launch_bounds__(256)
void bn_stats_node(const float* __restrict__ agg, const float* __restrict__ ob,
                   float* __restrict__ xout, float* __restrict__ sums, int nrows)
{
    const int tid = blockIdx.x * blockDim.x + threadIdx.x;
    const int c = tid & (HD - 1);
    const int rstart  = tid >> 7;
    const int rstride = (gridDim.x * blockDim.x) >> 7;
    const float bias = ob[c];
    float s1 = 0.0f, s2 = 0.0f;
    for (int r = rstart; r < nrows; r += rstride) {
        float v = agg[(size_t)r * HD + c] + bias;
        v = (v > 0.0f) ? v : 0.0f;
        xout[(size_t)r * HD + c] = v;
        s1 += v;
        s2 += v * v;
    }
    atomicAdd(&sums[c], s1);
    atomicAdd(&sums[HD + c], s2);
}

__global__ void bn_finalize_node(const float* __restrict__ sums,
                                 const float* __restrict__ g,
                                 const float* __restrict__ b,
                                 float* __restrict__ scsh, float inv_n)
{
    const int c = threadIdx.x;                   // 128 threads
    const float mu  = sums[c] * inv_n;
    const float var = sums[HD + c] * inv_n - mu * mu;
    const float sc  = g[c] * rsqrtf(var + 1e-5f);
    scsh[c]      = sc;
    scsh[HD + c] = b[c] - mu * sc;
}

// ---------------------------------------------------------------------------
// Apply BN in place, and sum-pool into p128[G,128] exploiting sorted seg:
// run-length accumulate per thread, flush on segment change (~few atomics/blk).
// 250 blocks x 128 threads, 200 rows per block.
// ---------------------------------------------------------------------------
#define POOL_ROWS 200
__global__ __launch_bounds__(128)
void bn_apply_pool(float* __restrict__ x, const float* __restrict__ scsh,
                   const int* __restrict__ seg, float* __restrict__ p128,
                   int nrows)
{
    const int c  = threadIdx.x;
    const int r0 = blockIdx.x * POOL_ROWS;
    const int r1 = min(r0 + POOL_ROWS, nrows);
    if (r0 >= r1) return;
    const float sc = scsh[c], sh = scsh[HD + c];
    int curseg = seg[r0];
    float accum = 0.0f;
    for (int r = r0; r < r1; ++r) {
        const int sg = seg[r];
        float v = x[(size_t)r * HD + c] * sc + sh;
        x[(size_t)r * HD + c] = v;
        if (sg != curseg) {
            atomicAdd(&p128[(size_t)curseg * HD + c], accum);
            accum = 0.0f;
            curseg = sg;
        }
        accum += v;
    }
    atomicAdd(&p128[(size_t)curseg * HD + c], accum);
}

// ---------------------------------------------------------------------------
// p = relu(p128 @ lw + lb); BN over G=64 graphs; write ph[:, layer*32 .. +32).
// Single block.
// ---------------------------------------------------------------------------
__global__ __launch_bounds__(256)
void graph_head(const float* __restrict__ p128, const float* __restrict__ lw,
                const float* __restrict__ lb, const float* __restrict__ lg,
                const float* __restrict__ lbt, float* __restrict__ ph, int layer)
{
    __shared__ float P[NGRAPH * HID];
    for (int idx = threadIdx.x; idx < NGRAPH * HID; idx += blockDim.x) {
        const int g = idx >> 5, j = idx & 31;
        float acc = lb[j];
        for (int k = 0; k < HD; ++k) acc += p128[(size_t)g * HD + k] * lw[k * HID + j];
        P[idx] = (acc > 0.0f) ? acc : 0.0f;
    }
    __syncthreads();
    if (threadIdx.x < HID) {
        const int j = threadIdx.x;
        float s1 = 0.0f, s2 = 0.0f;
        for (int g = 0; g < NGRAPH; ++g) { const float v = P[g * HID + j]; s1 += v; s2 += v * v; }
        const float mu  = s1 / (float)NGRAPH;
        const float var = s2 / (float)NGRAPH - mu * mu;
        const float sc  = lg[j] * rsqrtf(var + 1e-5f);
        const float sh  = lbt[j] - mu * sc;
        for (int g = 0; g < NGRAPH; ++g)
            ph[g * (3 * HID) + layer * HID + j] = P[g * HID + j] * sc + sh;
    }
}

// ---------------------------------------------------------------------------
// hmid = relu(BN(ph @ mw1)); logits = hmid @ mw2; out = log_softmax(logits).
// Single block.
// ---------------------------------------------------------------------------
__global__ __launch_bounds__(256)
void mlp_head(const float* __restrict__ ph, const float* __restrict__ mw1,
              const float* __restrict__ mg, const float* __restrict__ mb,
              const float* __restrict__ mw2, float* __restrict__ out)
{
    __shared__ float Hm[NGRAPH * HID];
    __shared__ float sc_s[HID], sh_s[HID];
    for (int idx = threadIdx.x; idx < NGRAPH * HID; idx += blockDim.x) {
        const int g = idx >> 5, j = idx & 31;
        float acc = 0.0f;
        for (int k = 0; k < 3 * HID; ++k) acc += ph[g * (3 * HID) + k] * mw1[k * HID + j];
        Hm[idx] = acc;
    }
    __syncthreads();
    if (threadIdx.x < HID) {
        const int j = threadIdx.x;
        float s1 = 0.0f, s2 = 0.0f;
        for (int g = 0; g < NGRAPH; ++g) { const float v = Hm[g * HID + j]; s1 += v; s2 += v * v; }
        const float mu  = s1 / (float)NGRAPH;
        const float var = s2 / (float)NGRAPH - mu * mu;
        sc_s[j] = mg[j] * rsqrtf(var + 1e-5f);
        sh_s[j] = mb[j] - mu * sc_s[j];
    }
    __syncthreads();
    for (int idx = threadIdx.x; idx < NGRAPH * HID; idx += blockDim.x) {
        const int j = idx & 31;
        float v = Hm[idx] * sc_s[j] + sh_s[j];
        Hm[idx] = (v > 0.0f) ? v : 0.0f;
    }
    __syncthreads();
    if (threadIdx.x < NGRAPH) {
        const int g = threadIdx.x;
        float logit[NOUT];
        float mx = -1e30f;
        for (int o = 0; o < NOUT; ++o) {
            float acc = 0.0f;
            for (int k = 0; k < HID; ++k) acc += Hm[g * HID + k] * mw2[k * NOUT + o];
            logit[o] = acc;
            mx = fmaxf(mx, acc);
        }
        float s = 0.0f;
        for (int o = 0; o < NOUT; ++o) s += __expf(logit[o] - mx);
        const float lse = mx + __logf(s);
        for (int o = 0; o < NOUT; ++o) out[g * NOUT + o] = logit[o] - lse;
    }
}

// ---------------------------------------------------------------------------
extern "C" void kernel_launch(void* const* d_in, const int* in_sizes, int n_in,
                              void* d_out, int out_size, void* d_ws, size_t ws_size,
                              hipStream_t stream)
{
    const float* feat = (const float*)d_in[0];
    const int*   src  = (const int*)d_in[1];
    const int*   dst  = (const int*)d_in[2];
    const int*   seg  = (const int*)d_in[3];
    const float* mw1  = (const float*)d_in[34];
    const float* mg   = (const float*)d_in[35];
    const float* mb   = (const float*)d_in[36];
    const float* mw2  = (const float*)d_in[37];
    float* out = (float*)d_out;

    // ---- workspace carve-out (256B aligned) ----
    char* ws = (char*)d_ws;
    size_t off = 0;
    auto carve = [&](size_t bytes) -> char* {
        char* p = ws + off;
        off = (off + bytes + 255) & ~(size_t)255;
        return p;
    };
    float* hbuf = (float*)carve((size_t)NNODES * HD * sizeof(float));   // 25.6 MB
    float* xbuf = (float*)carve((size_t)NNODES * HD * sizeof(float));   // 25.6 MB
    float* agg  = (float*)carve((size_t)NNODES * HD * sizeof(float));   // 25.6 MB
    float* ex   = (float*)carve((size_t)NEDGES * NHEAD * sizeof(float));// 12.8 MB
    float* den  = (float*)carve((size_t)NNODES * NHEAD * sizeof(float));// 0.8 MB
    float* sums = (float*)carve(2 * HD * sizeof(float));
    float* scsh = (float*)carve(2 * HD * sizeof(float));
    float* p128 = (float*)carve((size_t)NGRAPH * HD * sizeof(float));
    float* ph   = (float*)carve((size_t)NGRAPH * 3 * HID * sizeof(float));

    const int gemm_blocks = ((NNODES / 16) + 7) / 8;      // 391
    const int edge_blocks = (NEDGES * 32) / 256;          // 100000 (exact)
    const float inv_n = 1.0f / (float)NNODES;

    const float* xin = feat;
    for (int layer = 0; layer < 3; ++layer) {
        const int base = 4 + layer * 10;
        const float* W   = (const float*)d_in[base + 0];
        const float* fb  = (const float*)d_in[base + 1];
        const float* at  = (const float*)d_in[base + 2];
        const float* ob  = (const float*)d_in[base + 3];
        const float* bg  = (const float*)d_in[base + 4];
        const float* bb  = (const float*)d_in[base + 5];
        const float* lw  = (const float*)d_in[base + 6];
        const float* lb  = (const float*)d_in[base + 7];
        const float* lg  = (const float*)d_in[base + 8];
        const float* lbt = (const float*)d_in[base + 9];

        hipMemsetAsync(den, 0, (size_t)NNODES * NHEAD * sizeof(float), stream);
        hipMemsetAsync(agg, 0, (size_t)NNODES * HD * sizeof(float), stream);
        hipMemsetAsync(sums, 0, 2 * HD * sizeof(float), stream);
        hipMemsetAsync(p128, 0, (size_t)NGRAPH * HD * sizeof(float), stream);

        gemm_node_wmma<<<gemm_blocks, 256, 0, stream>>>(xin, W, fb, hbuf, NNODES);
        edge_scores<<<edge_blocks, 256, 0, stream>>>(hbuf, src, dst, at, ex, den, NEDGES);
        edge_aggregate<<<edge_blocks, 256, 0, stream>>>(hbuf, src, dst, ex, den, agg, NEDGES);
        bn_stats_node<<<256, 256, 0, stream>>>(agg, ob, xbuf, sums, NNODES);
        bn_finalize_node<<<1, 128, 0, stream>>>(sums, bg, bb, scsh, inv_n);
        bn_apply_pool<<<(NNODES + POOL_ROWS - 1) / POOL_ROWS, 128, 0, stream>>>(
            xbuf, scsh, seg, p128, NNODES);
        graph_head<<<1, 256, 0, stream>>>(p128, lw, lb, lg, lbt, ph, layer);

        xin = xbuf;   // next layer consumes normalized node features
    }

    mlp_head<<<1, 256, 0, stream>>>(ph, mw1, mg, mb, mw2, out);
}